// ConvTrace_15633680958282
// MI455X (gfx1250) — compile-verified
//
#include <hip/hip_runtime.h>
#include <hip/hip_bf16.h>

typedef __attribute__((ext_vector_type(2))) float v2f;
typedef __attribute__((ext_vector_type(8))) float v8f;
typedef int vi4 __attribute__((vector_size(16)));
typedef vi4 __attribute__((address_space(1)))* gvi4p;   // global int4*
typedef vi4 __attribute__((address_space(3)))* svi4p;   // LDS int4*

#define GSZ   256
#define HOUT  251
#define NMAT  512
#define LDP   260          // padded LDS row pitch (floats): stride-4 banks -> conflict-free A loads
#define MATEL 65536        // 256*256

__device__ __forceinline__ v8f wmma_f32(v2f a, v2f b, v8f c) {
  // V_WMMA_F32_16X16X4_F32 : D = A(16x4) * B(4x16) + C(16x16)
  return __builtin_amdgcn_wmma_f32_16x16x4_f32(false, a, false, b, (short)0, c, false, false);
}

__device__ __forceinline__ float wave_reduce(float v) {
  #pragma unroll
  for (int off = 16; off; off >>= 1) v += __shfl_xor(v, off, 32);
  return v;
}

__device__ __forceinline__ void async_wait0() {
#if __has_builtin(__builtin_amdgcn_s_wait_asynccnt)
  __builtin_amdgcn_s_wait_asynccnt(0);
#else
  asm volatile("s_wait_asynccnt 0" ::: "memory");
#endif
}

// Stage one 256x256 f32 matrix from global into padded LDS rows.
// Prefer the CDNA5 async memory->LDS path (GLOBAL_LOAD_ASYNC_TO_LDS_B128, ASYNCcnt).
__device__ __forceinline__ void stage_to_lds(const float4* __restrict__ src,
                                             float* lds, int tid) {
#if __has_builtin(__builtin_amdgcn_global_load_async_to_lds_b128)
  for (int i = tid; i < MATEL / 4; i += 256) {
    const int r = i >> 6, c = (i & 63) << 2;
    __builtin_amdgcn_global_load_async_to_lds_b128(
        (gvi4p)(src + i), (svi4p)&lds[r * LDP + c], 0, 0);
  }
  async_wait0();
#else
  for (int i = tid; i < MATEL / 4; i += 256) {
    float4 v = src[i];
    const int r = i >> 6, c = (i & 63) << 2;
    *(float4*)&lds[r * LDP + c] = v;
  }
#endif
}

// ---------------- conv: y = conv2d(x, w) + b, zero-padded to 256x256 ----------------
__global__ void __launch_bounds__(256) conv_kernel(const float* __restrict__ x,
                                                   const float* __restrict__ cw,
                                                   const float* __restrict__ cb,
                                                   float* __restrict__ Cg,
                                                   int matStart) {
  __shared__ float sw[36];
  __shared__ float sx[6][256];
  const int tid  = threadIdx.x;
  const int mloc = blockIdx.x >> 8;
  const int row  = blockIdx.x & 255;
  const int mat  = matStart + mloc;
  const int b    = mat >> 3;
  const int ch   = mat & 7;
  if (tid < 36) sw[tid] = cw[ch * 36 + tid];
  if (row <= HOUT - 1) {
    const float* xb = x + ((size_t)b << 16) + (size_t)row * GSZ;
    #pragma unroll
    for (int u = 0; u < 6; u++) sx[u][tid] = xb[u * GSZ + tid];
  }
  __syncthreads();
  float acc = 0.0f;
  const int col = tid;
  if (row <= HOUT - 1 && col <= HOUT - 1) {
    acc = cb[ch];
    #pragma unroll
    for (int u = 0; u < 6; u++)
      #pragma unroll
      for (int v = 0; v < 6; v++)
        acc = fmaf(sw[u * 6 + v], sx[u][col + v], acc);
  }
  Cg[((size_t)mloc << 16) + (size_t)row * GSZ + col] = acc;
}

// ---------------- pass 1: D = C*C (WMMA f32), tr(C^2) ----------------
__global__ void __launch_bounds__(256) square_kernel(const float* __restrict__ Cg,
                                                     float* __restrict__ Dg,
                                                     float* __restrict__ traces,
                                                     int matStart) {
  __shared__ float lds[GSZ * LDP];
  __shared__ float red[8];
  const int tid = threadIdx.x, wave = tid >> 5, lane = tid & 31;
  const int ln = lane & 15, hi = lane >> 4, kk = hi * 2;

  stage_to_lds((const float4*)(Cg + ((size_t)blockIdx.x << 16)), lds, tid);
  __syncthreads();

  // tr(C^2) = <C, C^T>
  float t2 = 0.0f;
  for (int i = tid; i < MATEL; i += 256) {
    int r = i >> 8, c = i & 255;
    t2 = fmaf(lds[r * LDP + c], lds[c * LDP + r], t2);
  }
  t2 = wave_reduce(t2);
  if (lane == 0) red[wave] = t2;
  __syncthreads();
  if (tid == 0) {
    float s = 0.0f;
    #pragma unroll
    for (int w = 0; w < 8; w++) s += red[w];
    traces[((matStart + blockIdx.x) << 2) + 0] = s;
  }

  // D = C*C : each wave owns a fixed 32-wide column band, 64x32 macro-tiles
  float* Dm = Dg + ((size_t)blockIdx.x << 16);
  const int C0 = wave << 5;
  for (int rr = 0; rr < 4; rr++) {
    const int R0 = rr << 6;
    v8f acc[4][2];
    #pragma unroll
    for (int a = 0; a < 4; a++) { acc[a][0] = (v8f){}; acc[a][1] = (v8f){}; }
    #pragma unroll 2
    for (int k = 0; k < GSZ; k += 4) {
      const int kr = k + kk;
      v2f A[4];
      #pragma unroll
      for (int a = 0; a < 4; a++)
        A[a] = *(const v2f*)&lds[(R0 + 16 * a + ln) * LDP + kr];
      v2f B0, B1;
      B0.x = lds[kr * LDP + C0 + ln];        B0.y = lds[(kr + 1) * LDP + C0 + ln];
      B1.x = lds[kr * LDP + C0 + 16 + ln];   B1.y = lds[(kr + 1) * LDP + C0 + 16 + ln];
      #pragma unroll
      for (int a = 0; a < 4; a++) {
        acc[a][0] = wmma_f32(A[a], B0, acc[a][0]);
        acc[a][1] = wmma_f32(A[a], B1, acc[a][1]);
      }
    }
    #pragma unroll
    for (int a = 0; a < 4; a++)
      #pragma unroll
      for (int r = 0; r < 8; r++) {
        const int m0 = R0 + 16 * a + hi * 8 + r;
        Dm[(size_t)m0 * GSZ + C0 + ln]      = acc[a][0][r];
        Dm[(size_t)m0 * GSZ + C0 + 16 + ln] = acc[a][1][r];
      }
  }
}

// ---------------- pass 2: F = C*D (= C^3) tiles, tr(C^3), tr(C^4), tr(C^5) ----------------
__global__ void __launch_bounds__(256) cube_kernel(const float* __restrict__ Cg,
                                                   const float* __restrict__ Dg,
                                                   float* __restrict__ traces,
                                                   int matStart) {
  __shared__ float lds[GSZ * LDP];
  __shared__ float red3[8], red4[8], red5[8];
  const int tid = threadIdx.x, wave = tid >> 5, lane = tid & 31;
  const int ln = lane & 15, hi = lane >> 4, kk = hi * 2;

  stage_to_lds((const float4*)(Cg + ((size_t)blockIdx.x << 16)), lds, tid);
  __syncthreads();

  const float* Dm = Dg + ((size_t)blockIdx.x << 16);

  // tr(C^4) = <D, D^T>
  float t4 = 0.0f;
  for (int i = tid; i < MATEL; i += 256) {
    int r = i >> 8, c = i & 255;
    t4 = fmaf(Dm[(size_t)r * GSZ + c], Dm[(size_t)c * GSZ + r], t4);
  }

  float t3 = 0.0f, t5 = 0.0f;
  const int C0 = wave << 5;
  for (int rr = 0; rr < 4; rr++) {
    const int R0 = rr << 6;
    v8f acc[4][2];
    #pragma unroll
    for (int a = 0; a < 4; a++) { acc[a][0] = (v8f){}; acc[a][1] = (v8f){}; }
    #pragma unroll 2
    for (int k = 0; k < GSZ; k += 4) {
      const int kr = k + kk;
      v2f A[4];
      #pragma unroll
      for (int a = 0; a < 4; a++)
        A[a] = *(const v2f*)&lds[(R0 + 16 * a + ln) * LDP + kr];
      v2f B0, B1;
      B0.x = Dm[(size_t)kr * GSZ + C0 + ln];        B0.y = Dm[(size_t)(kr + 1) * GSZ + C0 + ln];
      B1.x = Dm[(size_t)kr * GSZ + C0 + 16 + ln];   B1.y = Dm[(size_t)(kr + 1) * GSZ + C0 + 16 + ln];
      #pragma unroll
      for (int a = 0; a < 4; a++) {
        acc[a][0] = wmma_f32(A[a], B0, acc[a][0]);
        acc[a][1] = wmma_f32(A[a], B1, acc[a][1]);
      }
    }
    // contract F tile against D^T (tr C^5) and diagonal (tr C^3); F never hits memory
    #pragma unroll
    for (int a = 0; a < 4; a++)
      #pragma unroll
      for (int r = 0; r < 8; r++) {
        const int m0 = R0 + 16 * a + hi * 8 + r;
        const int n0 = C0 + ln;
        float f;
        f = acc[a][0][r]; t5 = fmaf(f, Dm[(size_t)n0 * GSZ + m0], t5);        if (m0 == n0)      t3 += f;
        f = acc[a][1][r]; t5 = fmaf(f, Dm[(size_t)(n0 + 16) * GSZ + m0], t5); if (m0 == n0 + 16) t3 += f;
      }
  }

  t3 = wave_reduce(t3);
  t4 = wave_reduce(t4);
  t5 = wave_reduce(t5);
  if (lane == 0) { red3[wave] = t3; red4[wave] = t4; red5[wave] = t5; }
  __syncthreads();
  if (tid == 0) {
    float s3 = 0, s4 = 0, s5 = 0;
    #pragma unroll
    for (int w = 0; w < 8; w++) { s3 += red3[w]; s4 += red4[w]; s5 += red5[w]; }
    const int base = (matStart + blockIdx.x) << 2;
    traces[base + 1] = s3;
    traces[base + 2] = s4;
    traces[base + 3] = s5;
  }
}

// ---------------- finalize: out[b] = sum_{ch,i,j} coef * ts^(j+1) / numel^(i+j+1) ----------------
__global__ void finalize_kernel(const float* __restrict__ traces,
                                const float* __restrict__ coef,
                                float* __restrict__ out) {
  const int b = threadIdx.x;
  if (b >= 64) return;
  const float numel = (float)(HOUT * HOUT);
  float s = 0.0f;
  for (int ch = 0; ch < 8; ch++) {
    for (int i = 0; i < 4; i++) {
      const float ts = traces[((b * 8 + ch) << 2) + i];
      float d = 1.0f;
      for (int e = 0; e <= i; e++) d *= numel;   // numel^(i+1)
      float tp = 1.0f;
      for (int j = 0; j < 3; j++) {
        tp *= ts;                                 // ts^(j+1)
        s += coef[ch * 12 + i * 3 + j] * tp / d;  // / numel^(i+j+1)
        d *= numel;
      }
    }
  }
  out[b] = s;
}

extern "C" void kernel_launch(void* const* d_in, const int* in_sizes, int n_in,
                              void* d_out, int out_size, void* d_ws, size_t ws_size,
                              hipStream_t stream) {
  const float* x    = (const float*)d_in[0];   // [64,256,256]
  const float* cw   = (const float*)d_in[1];   // [8,1,6,6]
  const float* cb   = (const float*)d_in[2];   // [8]
  const float* coef = (const float*)d_in[3];   // [8,4,3]
  float* out = (float*)d_out;

  char* ws = (char*)d_ws;
  float* traces = (float*)ws;                          // 512*4 floats, fully overwritten
  size_t off = ((size_t)NMAT * 4 * sizeof(float) + 255) & ~(size_t)255;
  const size_t perMat = (size_t)MATEL * sizeof(float); // 256 KB
  size_t avail = (ws_size > off) ? ws_size - off : 0;
  int chunk = (int)(avail / (2 * perMat));
  if (chunk < 1) chunk = 1;
  if (chunk > NMAT) chunk = NMAT;

  float* Cgp = (float*)(ws + off);
  float* Dgp = (float*)(ws + off + (size_t)chunk * perMat);

  for (int m0 = 0; m0 < NMAT; m0 += chunk) {
    int c = (NMAT - m0 < chunk) ? (NMAT - m0) : chunk;
    conv_kernel  <<<c * 256, 256, 0, stream>>>(x, cw, cb, Cgp, m0);
    square_kernel<<<c,       256, 0, stream>>>(Cgp, Dgp, traces, m0);
    cube_kernel  <<<c,       256, 0, stream>>>(Cgp, Dgp, traces, m0);
  }
  finalize_kernel<<<1, 64, 0, stream>>>(traces, coef, out);
}